// GraphEaseeNet_5128190951974
// MI455X (gfx1250) — compile-verified
//
#include <hip/hip_runtime.h>
#include <climits>

typedef __attribute__((ext_vector_type(2))) float v2f;
typedef __attribute__((ext_vector_type(4))) float v4f;
typedef __attribute__((ext_vector_type(8))) float v8f;
typedef __attribute__((ext_vector_type(4))) int   v4i;

#define CCH 128
#define EPSBN 1e-5f
#define A1_STR 260   // padded LDS row stride for 256-wide activations (bank-conflict free)
#define A2_STR 132   // padded LDS row stride for 128-wide activations

__device__ __forceinline__ v8f zero8() {
    v8f z;
    #pragma unroll
    for (int i = 0; i < 8; ++i) z[i] = 0.0f;
    return z;
}

// order-preserving float <-> signed-int map (for atomicMax-based segment_max)
__device__ __forceinline__ int f2ord(float f) {
    int b = __float_as_int(f);
    return b >= 0 ? b : (b ^ 0x7fffffff);
}
__device__ __forceinline__ float ord2f(int o) {
    int b = o >= 0 ? o : (o ^ 0x7fffffff);
    return __int_as_float(b);
}

// ---------------------------------------------------------------------------
// Layer 0: g0 = relu(bn0( [x_a @ w0a.T + b0a ; x_v @ w0v.T + b0v] ))
// One wave computes a 16-node x 128-channel tile with f32 WMMA.
// ---------------------------------------------------------------------------
__global__ void __launch_bounds__(128)
layer0_kernel(const float* __restrict__ x,
              const float* __restrict__ w0a, const float* __restrict__ b0a,
              const float* __restrict__ w0v, const float* __restrict__ b0v,
              const float* __restrict__ bn0,
              const int* __restrict__ pNA,
              float* __restrict__ g0, int N)
{
    const int wave = threadIdx.x >> 5;
    const int lane = threadIdx.x & 31;
    const int lrow = lane & 15;
    const int hi   = lane >> 4;
    const int tile = blockIdx.x * (blockDim.x >> 5) + wave;
    const int ntiles = N >> 4;
    if (tile >= ntiles) return;

    const int NA   = *pNA;                 // 16-divisible: tiles are purely audio or video
    const int row0 = tile * 16;
    // wave-uniform selection: force scalar so the k-loop is non-divergent
    const int au   = __builtin_amdgcn_readfirstlane((row0 < NA) ? 1 : 0);
    const float* W    = au ? w0a : w0v;
    const float* bias = au ? b0a : b0v;
    const int kmax = au ? 1024 : 512;
    const int wstr = kmax;
    const int soff = au ? 0 : 1024;        // x[:,0,:1024] vs x[:,1,:512]

    const float* aP = x + (size_t)(row0 + lrow) * 2048 + soff + hi * 2;
    const float* bP0 = W + (size_t)lrow * wstr + hi * 2;
    const float* bPn[8];
    #pragma unroll
    for (int nt = 0; nt < 8; ++nt) bPn[nt] = bP0 + (size_t)nt * 16 * wstr;

    v8f acc[8];
    #pragma unroll
    for (int nt = 0; nt < 8; ++nt) acc[nt] = zero8();

    #pragma unroll 4
    for (int k = 0; k < kmax; k += 4) {
        v2f a = *(const v2f*)(aP + k);
        #pragma unroll
        for (int nt = 0; nt < 8; ++nt) {
            v2f b = *(const v2f*)(bPn[nt] + k);
            acc[nt] = __builtin_amdgcn_wmma_f32_16x16x4_f32(
                false, a, false, b, (short)0, acc[nt], false, false);
        }
    }

    #pragma unroll
    for (int nt = 0; nt < 8; ++nt) {
        const int col = nt * 16 + lrow;
        const float g = bn0[col], bb = bn0[CCH + col];
        const float m = bn0[2 * CCH + col], v = bn0[3 * CCH + col];
        const float s = g * rsqrtf(v + EPSBN);
        const float t = bb - m * s + bias[col] * s;
        #pragma unroll
        for (int r = 0; r < 8; ++r) {
            const float val = fmaxf(acc[nt][r] * s + t, 0.0f);
            const int rowAbs = row0 + r + hi * 8;
            g0[(size_t)rowAbs * CCH + col] = val;
        }
    }
}

// ---------------------------------------------------------------------------
__global__ void fill_kernel(v4i* __restrict__ p, int val, int n4)
{
    int i = blockIdx.x * blockDim.x + threadIdx.x;
    if (i < n4) {
        v4i v; v[0] = val; v[1] = val; v[2] = val; v[3] = val;
        p[i] = v;
    }
}

// ---------------------------------------------------------------------------
// EdgeConv: msg = LPP([x_i ; x_j - x_i]); atomic segment-max into agg (ordered-int).
// 64-thread block = 2 waves, 32 edges; both GEMMs with f32 WMMA.
// ---------------------------------------------------------------------------
__global__ void __launch_bounds__(64)
edgeconv_kernel(const float* __restrict__ feats,
                const int* __restrict__ edge_index,
                const int* __restrict__ edge_delta,
                const int* __restrict__ edge_self,
                const float* __restrict__ bn1,   // [4,256] slice
                const float* __restrict__ W1,    // [128,256]
                const float* __restrict__ bn2,   // [4,128] slice
                const float* __restrict__ W2,    // [128,128]
                int* __restrict__ agg,
                int E, int masksel)
{
    __shared__ float act [32 * A1_STR];
    __shared__ float act2[32 * A2_STR];
    __shared__ int   sdst[32];
    __shared__ int   sact[32];

    const int tid = threadIdx.x;
    const int e0  = blockIdx.x * 32;

    // ---------------- Phase 1: gather + BN1 + ReLU into LDS ----------------
    {
        const int el   = tid >> 1;     // edge within tile
        const int half = tid & 1;      // channel half [0,64) or [64,128)
        const int e    = e0 + el;
        int srcn = 0, dstn = 0, activ = 0;
        if (e < E) {
            srcn = edge_index[e];
            dstn = edge_index[E + e];
            const int d  = edge_delta[e];
            const int sf = edge_self[e];
            activ = masksel ? (((d >= 1) && (d < 4)) || (sf == 1)) : (d < 1);
        }
        if (half == 0) { sdst[el] = dstn; sact[el] = activ; }
        const float* xi = feats + (size_t)dstn * CCH;
        const float* xj = feats + (size_t)srcn * CCH;
        const int cbase = half * 64;
        for (int c4 = 0; c4 < 64; c4 += 4) {
            const int c = cbase + c4;
            v4f vi, vj;
            if (e < E) { vi = *(const v4f*)(xi + c); vj = *(const v4f*)(xj + c); }
            else {
                #pragma unroll
                for (int j = 0; j < 4; ++j) { vi[j] = 0.0f; vj[j] = 0.0f; }
            }
            // channels [c, c+4) of concat part 0 (x_i)
            const v4f gA = *(const v4f*)(bn1 + c);
            const v4f bA = *(const v4f*)(bn1 + 256 + c);
            const v4f mA = *(const v4f*)(bn1 + 512 + c);
            const v4f vA = *(const v4f*)(bn1 + 768 + c);
            // channels [128+c, 128+c+4) of concat part 1 (x_j - x_i)
            const v4f gB = *(const v4f*)(bn1 + 128 + c);
            const v4f bB = *(const v4f*)(bn1 + 256 + 128 + c);
            const v4f mB = *(const v4f*)(bn1 + 512 + 128 + c);
            const v4f vB = *(const v4f*)(bn1 + 768 + 128 + c);
            #pragma unroll
            for (int j = 0; j < 4; ++j) {
                const float sA = gA[j] * rsqrtf(vA[j] + EPSBN);
                const float tA = bA[j] - mA[j] * sA;
                act[el * A1_STR + c + j] = fmaxf(vi[j] * sA + tA, 0.0f);
                const float sB = gB[j] * rsqrtf(vB[j] + EPSBN);
                const float tB = bB[j] - mB[j] * sB;
                act[el * A1_STR + 128 + c + j] = fmaxf((vj[j] - vi[j]) * sB + tB, 0.0f);
            }
        }
    }
    __syncthreads();

    const int wave = tid >> 5;
    const int lane = tid & 31;
    const int lrow = lane & 15;
    const int hi   = lane >> 4;

    // ---------------- GEMM1: [16 x 256] @ W1.T -> [16 x 128] ----------------
    {
        const float* aP = act + (wave * 16 + lrow) * A1_STR + hi * 2;
        const float* bP0 = W1 + lrow * 256 + hi * 2;
        const float* bPn[8];
        #pragma unroll
        for (int nt = 0; nt < 8; ++nt) bPn[nt] = bP0 + nt * 16 * 256;

        v8f acc[8];
        #pragma unroll
        for (int nt = 0; nt < 8; ++nt) acc[nt] = zero8();

        #pragma unroll 4
        for (int k = 0; k < 256; k += 4) {
            v2f a = *(const v2f*)(aP + k);
            #pragma unroll
            for (int nt = 0; nt < 8; ++nt) {
                v2f b = *(const v2f*)(bPn[nt] + k);
                acc[nt] = __builtin_amdgcn_wmma_f32_16x16x4_f32(
                    false, a, false, b, (short)0, acc[nt], false, false);
            }
        }
        // BN2 + ReLU, restage as A-matrix for GEMM2
        #pragma unroll
        for (int nt = 0; nt < 8; ++nt) {
            const int col = nt * 16 + lrow;
            const float g = bn2[col], bb = bn2[CCH + col];
            const float m = bn2[2 * CCH + col], v = bn2[3 * CCH + col];
            const float s = g * rsqrtf(v + EPSBN);
            const float t = bb - m * s;
            #pragma unroll
            for (int r = 0; r < 8; ++r) {
                const float val = fmaxf(acc[nt][r] * s + t, 0.0f);
                act2[(wave * 16 + r + hi * 8) * A2_STR + col] = val;
            }
        }
    }
    __syncthreads();

    // ---------------- GEMM2: [16 x 128] @ W2.T -> [16 x 128] ----------------
    {
        const float* aP = act2 + (wave * 16 + lrow) * A2_STR + hi * 2;
        const float* bP0 = W2 + lrow * 128 + hi * 2;
        const float* bPn[8];
        #pragma unroll
        for (int nt = 0; nt < 8; ++nt) bPn[nt] = bP0 + nt * 16 * 128;

        v8f acc[8];
        #pragma unroll
        for (int nt = 0; nt < 8; ++nt) acc[nt] = zero8();

        #pragma unroll 4
        for (int k = 0; k < 128; k += 4) {
            v2f a = *(const v2f*)(aP + k);
            #pragma unroll
            for (int nt = 0; nt < 8; ++nt) {
                v2f b = *(const v2f*)(bPn[nt] + k);
                acc[nt] = __builtin_amdgcn_wmma_f32_16x16x4_f32(
                    false, a, false, b, (short)0, acc[nt], false, false);
            }
        }
        // scatter: segment-max via ordered-int atomicMax
        #pragma unroll
        for (int r = 0; r < 8; ++r) {
            const int erow = wave * 16 + r + hi * 8;
            const int e = e0 + erow;
            if (e < E && sact[erow]) {
                int* base = agg + (size_t)sdst[erow] * CCH;
                #pragma unroll
                for (int nt = 0; nt < 8; ++nt) {
                    atomicMax(base + nt * 16 + lrow, f2ord(acc[nt][r]));
                }
            }
        }
    }
}

// ---------------------------------------------------------------------------
// Finalize: resolve max (isolated -> 0), then per layer position:
//   mode 0: raw    mode 1: relu(bn(.))    mode 2: relu(bn(. + res))    mode 3: . + res
// ---------------------------------------------------------------------------
__global__ void finalize_kernel(const int* __restrict__ agg,
                                const float* res,
                                const float* __restrict__ bn,
                                float* out, int total, int mode)
{
    const int i = blockIdx.x * blockDim.x + threadIdx.x;
    if (i >= total) return;
    const int o = agg[i];
    float f = (o == INT_MIN) ? 0.0f : ord2f(o);
    if (mode >= 2) f += res[i];
    if (mode == 1 || mode == 2) {
        const int c = i & (CCH - 1);
        const float g = bn[c], bb = bn[CCH + c];
        const float m = bn[2 * CCH + c], v = bn[3 * CCH + c];
        const float s = g * rsqrtf(v + EPSBN);
        f = fmaxf((f - m) * s + bb, 0.0f);
    }
    out[i] = f;
}

// ---------------------------------------------------------------------------
// Heads: out = g4 @ w_heads[0].T + b ;  audio/video heads from g0.
// d_out layout: [N*2 main | NA*2 audio | (N-NA)*2 video]
// ---------------------------------------------------------------------------
__global__ void heads_kernel(const float* __restrict__ g4,
                             const float* __restrict__ g0,
                             const float* __restrict__ w_heads,  // [3,2,128]
                             const float* __restrict__ b_heads,  // [3,2]
                             const int* __restrict__ pNA,
                             float* __restrict__ out, int N)
{
    const int n = blockIdx.x * blockDim.x + threadIdx.x;
    if (n >= N) return;
    const int NA = *pNA;

    const float* r4 = g4 + (size_t)n * CCH;
    float a0 = 0.0f, a1 = 0.0f;
    for (int c = 0; c < CCH; c += 4) {
        const v4f xv = *(const v4f*)(r4 + c);
        const v4f w0 = *(const v4f*)(w_heads + c);
        const v4f w1 = *(const v4f*)(w_heads + CCH + c);
        #pragma unroll
        for (int j = 0; j < 4; ++j) { a0 += xv[j] * w0[j]; a1 += xv[j] * w1[j]; }
    }
    out[2 * n]     = a0 + b_heads[0];
    out[2 * n + 1] = a1 + b_heads[1];

    const int head = (n < NA) ? 1 : 2;
    const float* wh = w_heads + head * 2 * CCH;
    const float* r0 = g0 + (size_t)n * CCH;
    float c0 = 0.0f, c1 = 0.0f;
    for (int c = 0; c < CCH; c += 4) {
        const v4f xv = *(const v4f*)(r0 + c);
        const v4f w0 = *(const v4f*)(wh + c);
        const v4f w1 = *(const v4f*)(wh + CCH + c);
        #pragma unroll
        for (int j = 0; j < 4; ++j) { c0 += xv[j] * w0[j]; c1 += xv[j] * w1[j]; }
    }
    const int off = (n < NA) ? (2 * N + 2 * n) : (2 * N + 2 * NA + 2 * (n - NA));
    out[off]     = c0 + b_heads[head * 2];
    out[off + 1] = c1 + b_heads[head * 2 + 1];
}

// ---------------------------------------------------------------------------
extern "C" void kernel_launch(void* const* d_in, const int* in_sizes, int n_in,
                              void* d_out, int out_size, void* d_ws, size_t ws_size,
                              hipStream_t stream)
{
    const float* x          = (const float*)d_in[0];
    const int*   edge_index = (const int*)d_in[1];
    const int*   edge_delta = (const int*)d_in[2];
    const int*   edge_self  = (const int*)d_in[3];
    const int*   pNA        = (const int*)d_in[4];
    const float* w0a        = (const float*)d_in[5];
    const float* b0a        = (const float*)d_in[6];
    const float* w0v        = (const float*)d_in[7];
    const float* b0v        = (const float*)d_in[8];
    const float* bn0        = (const float*)d_in[9];
    const float* bn_node    = (const float*)d_in[10];
    const float* ec_bn1     = (const float*)d_in[11];
    const float* ec_fc1     = (const float*)d_in[12];
    const float* ec_bn2     = (const float*)d_in[13];
    const float* ec_fc2     = (const float*)d_in[14];
    const float* w_heads    = (const float*)d_in[15];
    const float* b_heads    = (const float*)d_in[16];

    const int N  = in_sizes[0] / 2048;   // x is [N,2,1024]
    const int E  = in_sizes[1] / 2;      // edge_index is [2,E]
    const int NC = N * CCH;

    float* buf0 = (float*)d_ws;          // g0 (kept for heads)
    float* buf1 = buf0 + NC;             // scratch d
    float* buf2 = buf1 + NC;             // current g
    int*   agg  = (int*)(buf2 + NC);     // max-aggregation (ordered ints)

    // layer 0
    {
        const int ntiles = (N + 15) / 16;
        const int blocks = (ntiles + 3) / 4;
        layer0_kernel<<<blocks, 128, 0, stream>>>(x, w0a, b0a, w0v, b0v, bn0, pNA, buf0, N);
    }

    const float* inb[8] = {buf0, buf1, buf2, buf1, buf2, buf1, buf2, buf1};
    float*      outb[8] = {buf1, buf2, buf1, buf2, buf1, buf2, buf1, buf2};
    const int  modes[8] = {0, 1, 0, 2, 0, 2, 0, 3};
    const float* bns[8] = {nullptr, bn_node, nullptr, bn_node + 4 * CCH,
                           nullptr, bn_node + 8 * CCH, nullptr, nullptr};
    const int   msel[8] = {0, 1, 0, 1, 0, 1, 0, 1};

    const int fillBlocks = (NC / 4 + 255) / 256;
    const int ecBlocks   = (E + 31) / 32;
    const int finBlocks  = (NC + 255) / 256;

    for (int L = 0; L < 8; ++L) {
        fill_kernel<<<fillBlocks, 256, 0, stream>>>((v4i*)agg, INT_MIN, NC / 4);
        edgeconv_kernel<<<ecBlocks, 64, 0, stream>>>(
            inb[L], edge_index, edge_delta, edge_self,
            ec_bn1 + (size_t)L * 4 * 256, ec_fc1 + (size_t)L * 128 * 256,
            ec_bn2 + (size_t)L * 4 * CCH, ec_fc2 + (size_t)L * CCH * CCH,
            agg, E, msel[L]);
        finalize_kernel<<<finBlocks, 256, 0, stream>>>(agg, outb[L], bns[L], outb[L], NC, modes[L]);
    }

    heads_kernel<<<(N + 255) / 256, 256, 0, stream>>>(
        buf2, buf0, w_heads, b_heads, pNA, (float*)d_out, N);
}